// HGWaveNet_57011395887442
// MI455X (gfx1250) — compile-verified
//
#include <hip/hip_runtime.h>
#include <hip/hip_bf16.h>
#include <math.h>

// ---------------------------------------------------------------------------
// HGWaveNet forward on MI455X (gfx1250, wave32, WMMA).
// All GEMMs use V_WMMA_F32_16X16X4_F32 (fp32 precision, matches reference).
// ---------------------------------------------------------------------------

#define DIM 128
#define EPSV 1e-10f

typedef float v2f __attribute__((ext_vector_type(2)));
typedef float v8f __attribute__((ext_vector_type(8)));

__device__ __forceinline__ float wave_reduce_sum(float v) {
    // wave32 butterfly reduction
    v += __shfl_xor(v, 16, 32);
    v += __shfl_xor(v, 8, 32);
    v += __shfl_xor(v, 4, 32);
    v += __shfl_xor(v, 2, 32);
    v += __shfl_xor(v, 1, 32);
    return v;
}

// -------- rowwise log_map(origin, x): t = (2/sc)*atanh(sc*||x||)/||x|| * x --
__global__ __launch_bounds__(256) void logmap0_kernel(
    const float* __restrict__ X, float* __restrict__ T,
    const float* __restrict__ curv, int N)
{
    const int row  = (int)((blockIdx.x * blockDim.x + threadIdx.x) >> 5);
    const int lane = threadIdx.x & 31;
    if (row >= N) return;
    const float4 v = reinterpret_cast<const float4*>(X + (size_t)row * DIM)[lane];
    float ss = v.x * v.x + v.y * v.y + v.z * v.z + v.w * v.w;
    ss = wave_reduce_sum(ss);
    const float c  = curv[0];
    const float sc = sqrtf(c);
    const float dn = fmaxf(sqrtf(ss), EPSV);
    const float s  = (2.0f / sc) * atanhf(sc * dn) / dn;
    float4 o; o.x = v.x * s; o.y = v.y * s; o.z = v.z * s; o.w = v.w * s;
    reinterpret_cast<float4*>(T + (size_t)row * DIM)[lane] = o;
}

// -------- Y = X @ W (+ addin), X:[N,128] fp32, W:[128,128] fp32 ------------
// 256 threads = 8 waves; each wave -> 16 rows x 128 cols (8 WMMA tiles).
// W pre-packed into LDS in fragment order: one ds_load_b64 per B fragment.
__global__ __launch_bounds__(256) void gemm128_kernel(
    const float* __restrict__ X, const float* __restrict__ W,
    const float* __restrict__ addin, float* __restrict__ Y, int N)
{
    __shared__ v2f wlds[32 * 8 * 32];   // [ks][nt][lane] -> 64 KB of 320 KB

    const int tid = threadIdx.x;
    // cooperative pack of W into WMMA-B fragment order
    for (int i = tid; i < 32 * 8 * 32; i += 256) {
        const int lane = i & 31;
        const int nt   = (i >> 5) & 7;
        const int ks   = i >> 8;
        const int k    = 4 * ks + 2 * (lane >> 4);   // B: VGPR0->K, VGPR1->K+1
        const int n    = 16 * nt + (lane & 15);      // B: lane indexes N
        v2f w;
        w.x = W[(size_t)k * DIM + n];
        w.y = W[(size_t)(k + 1) * DIM + n];
        wlds[i] = w;
    }
    __syncthreads();

    const int wave = tid >> 5;
    const int lane = tid & 31;
    const int m0   = ((int)blockIdx.x * 8 + wave) * 16;
    if (m0 >= N) return;                 // wave-uniform: EXEC stays all-1s

    const int mrow = m0 + (lane & 15);   // A: M = lane%16
    const int koff = 2 * (lane >> 4);    // A: lanes 16-31 hold K+2,K+3

    v8f acc[8];
    const v8f zero = {0.f, 0.f, 0.f, 0.f, 0.f, 0.f, 0.f, 0.f};
#pragma unroll
    for (int nt = 0; nt < 8; ++nt) acc[nt] = zero;

    const float* xrow = X + (size_t)mrow * DIM + koff;
#pragma unroll 4
    for (int ks = 0; ks < 32; ++ks) {
        const v2f a = *reinterpret_cast<const v2f*>(xrow + 4 * ks);
#pragma unroll
        for (int nt = 0; nt < 8; ++nt) {
            const v2f b = wlds[(ks * 8 + nt) * 32 + lane];
            acc[nt] = __builtin_amdgcn_wmma_f32_16x16x4_f32(
                false, a, false, b, (short)0, acc[nt], false, false);
        }
    }

    // epilogue: C/D layout -> element (M = r + 8*(lane>>4), N = 16*nt + lane%16)
    const int nlo = lane & 15;
    const int hi  = lane >> 4;
    const bool has_add = (addin != nullptr);
#pragma unroll
    for (int r = 0; r < 8; ++r) {
        const int m = m0 + r + 8 * hi;
        float* yrow = Y + (size_t)m * DIM;
        const float* arow = addin + (size_t)m * DIM;
#pragma unroll
        for (int nt = 0; nt < 8; ++nt) {
            float v = acc[nt][r];
            if (has_add) v += arow[nt * 16 + nlo];
            yrow[nt * 16 + nlo] = v;
        }
    }
}

// -------- edge scatter: msg[dst] += h[src]; deg[dst] += 1 ------------------
__global__ __launch_bounds__(256) void scatter_kernel(
    const float* __restrict__ h, const int* __restrict__ src,
    const int* __restrict__ dst, float* __restrict__ msg,
    float* __restrict__ deg, int E)
{
    const int e    = (int)((blockIdx.x * blockDim.x + threadIdx.x) >> 5);
    const int lane = threadIdx.x & 31;
    if (e >= E) return;
    const int s = src[e];
    const int d = dst[e];
    const float4 v = reinterpret_cast<const float4*>(h + (size_t)s * DIM)[lane];
    float* m = msg + (size_t)d * DIM + lane * 4;
    atomicAdd(m + 0, v.x);
    atomicAdd(m + 1, v.y);
    atomicAdd(m + 2, v.z);
    atomicAdd(m + 3, v.w);
    if (lane == 0) atomicAdd(deg + d, 1.0f);
}

// -------- fused: neigh = msg/deg; current = expmap0; tc = logmap0 ----------
__global__ __launch_bounds__(256) void tc_kernel(
    const float* __restrict__ msg, const float* __restrict__ deg,
    float* __restrict__ tc, const float* __restrict__ curv, int N)
{
    const int row  = (int)((blockIdx.x * blockDim.x + threadIdx.x) >> 5);
    const int lane = threadIdx.x & 31;
    if (row >= N) return;
    const float dg = fmaxf(deg[row], 1.0f);
    float4 v = reinterpret_cast<const float4*>(msg + (size_t)row * DIM)[lane];
    v.x /= dg; v.y /= dg; v.z /= dg; v.w /= dg;

    const float c  = curv[0];
    const float sc = sqrtf(c);

    float ss = wave_reduce_sum(v.x * v.x + v.y * v.y + v.z * v.z + v.w * v.w);
    const float vn = fmaxf(sqrtf(ss), EPSV);
    const float f  = tanhf(sc * vn * 0.5f) / (sc * vn);     // exp_map(0, .)
    float4 cur; cur.x = v.x * f; cur.y = v.y * f; cur.z = v.z * f; cur.w = v.w * f;

    float ss2 = wave_reduce_sum(cur.x * cur.x + cur.y * cur.y +
                                cur.z * cur.z + cur.w * cur.w);
    const float dn = fmaxf(sqrtf(ss2), EPSV);
    const float g  = (2.0f / sc) * atanhf(sc * dn) / dn;    // log_map(0, .)
    float4 o; o.x = cur.x * g; o.y = cur.y * g; o.z = cur.z * g; o.w = cur.w * g;
    reinterpret_cast<float4*>(tc + (size_t)row * DIM)[lane] = o;
}

// -------- final: out = expmap0(out) in place -------------------------------
__global__ __launch_bounds__(256) void expmap0_kernel(
    float* __restrict__ Y, const float* __restrict__ curv, int N)
{
    const int row  = (int)((blockIdx.x * blockDim.x + threadIdx.x) >> 5);
    const int lane = threadIdx.x & 31;
    if (row >= N) return;
    const float4 v = reinterpret_cast<const float4*>(Y + (size_t)row * DIM)[lane];
    float ss = wave_reduce_sum(v.x * v.x + v.y * v.y + v.z * v.z + v.w * v.w);
    const float c  = curv[0];
    const float sc = sqrtf(c);
    const float vn = fmaxf(sqrtf(ss), EPSV);
    const float f  = tanhf(sc * vn * 0.5f) / (sc * vn);
    float4 o; o.x = v.x * f; o.y = v.y * f; o.z = v.z * f; o.w = v.w * f;
    reinterpret_cast<float4*>(Y + (size_t)row * DIM)[lane] = o;
}

// ---------------------------------------------------------------------------
extern "C" void kernel_launch(void* const* d_in, const int* in_sizes, int n_in,
                              void* d_out, int out_size, void* d_ws, size_t ws_size,
                              hipStream_t stream)
{
    const float* node = (const float*)d_in[0];
    const float* gcw  = (const float*)d_in[1];
    const float* w0   = (const float*)d_in[2];
    const float* w1   = (const float*)d_in[3];
    const float* w2   = (const float*)d_in[4];
    const float* curv = (const float*)d_in[5];
    const int*   src  = (const int*)d_in[6];
    const int*   dst  = (const int*)d_in[7];
    const float* h0   = (const float*)d_in[8];
    const float* h1   = (const float*)d_in[9];
    const float* h2   = (const float*)d_in[10];
    float* out = (float*)d_out;

    const int N = in_sizes[0] / DIM;
    const int E = in_sizes[6];

    // workspace layout: t | h(/tc) | msg | deg   (~154 MB for N=100k)
    float* t   = (float*)d_ws;
    float* h   = t   + (size_t)N * DIM;
    float* msg = h   + (size_t)N * DIM;
    float* deg = msg + (size_t)N * DIM;

    const unsigned rowBlocks  = (unsigned)((N + 7) / 8);       // wave-per-row
    const unsigned gemmBlocks = (unsigned)((N + 127) / 128);   // 128 rows/block
    const unsigned edgeBlocks = (unsigned)((E + 7) / 8);       // wave-per-edge

    // 1) tangent of node embeddings, 2) GCN linear
    logmap0_kernel<<<rowBlocks, 256, 0, stream>>>(node, t, curv, N);
    gemm128_kernel<<<gemmBlocks, 256, 0, stream>>>(t, gcw, nullptr, h, N);

    // 3) mean aggregation over edges
    hipMemsetAsync(msg, 0, (size_t)N * DIM * sizeof(float), stream);
    hipMemsetAsync(deg, 0, (size_t)N * sizeof(float), stream);
    scatter_kernel<<<edgeBlocks, 256, 0, stream>>>(h, src, dst, msg, deg, E);

    // 4) tc = logmap0(expmap0(neigh))  (h is free now -> reuse as tc)
    tc_kernel<<<rowBlocks, 256, 0, stream>>>(msg, deg, h, curv, N);

    // 5) result = tc + logmap0(hist_i) @ w_i   (chained in place on d_out)
    logmap0_kernel<<<rowBlocks, 256, 0, stream>>>(h0, t, curv, N);
    gemm128_kernel<<<gemmBlocks, 256, 0, stream>>>(t, w0, h, out, N);
    logmap0_kernel<<<rowBlocks, 256, 0, stream>>>(h1, t, curv, N);
    gemm128_kernel<<<gemmBlocks, 256, 0, stream>>>(t, w1, out, out, N);
    logmap0_kernel<<<rowBlocks, 256, 0, stream>>>(h2, t, curv, N);
    gemm128_kernel<<<gemmBlocks, 256, 0, stream>>>(t, w2, out, out, N);

    // 6) final exp map
    expmap0_kernel<<<rowBlocks, 256, 0, stream>>>(out, curv, N);
}